// SimilarityLayer_25701084299578
// MI455X (gfx1250) — compile-verified
//
#include <hip/hip_runtime.h>

typedef float v2f __attribute__((ext_vector_type(2)));
typedef float v8f __attribute__((ext_vector_type(8)));

#define ROWS_PER_BLOCK 4096
#define TOPK 11          // top_n + 1 (self match dropped at the end)
#define MAXC 2816        // >= ceil(1e6/4096)*11 = 2695

__device__ __forceinline__ bool better(float v1, int i1, float v2, int i2) {
    return (v1 > v2) || (v1 == v2 && i1 < i2);
}

// ---------------------------------------------------------------------------
// Kernel A: fused dot/norm/similarity + per-block top-11.
// Each wave computes 16 rows per WMMA chain: D(16x16) += A(16x4) x B(4x16),
// with B = target chunk broadcast over N (so D[:,j] == dots for every j).
// ---------------------------------------------------------------------------
__global__ __launch_bounds__(256) void sim_topk_block_kernel(
    const float* __restrict__ emb, const int* __restrict__ user_id_p,
    int V, float* __restrict__ wsV, int* __restrict__ wsI)
{
    __shared__ float tgt[64];
    __shared__ float s_invt;
    __shared__ float sims[ROWS_PER_BLOCK];
    __shared__ float redV[256];
    __shared__ int   redI[256];

    const int tid = threadIdx.x;

    // Load target embedding + its inverse norm (tiny, once per block).
    if (tid < 64) tgt[tid] = emb[(size_t)(*user_id_p) * 64 + tid];
    __syncthreads();
    if (tid == 0) {
        float s = 0.f;
        for (int i = 0; i < 64; ++i) s = __builtin_fmaf(tgt[i], tgt[i], s);
        s_invt = rsqrtf(s);
    }
    __syncthreads();
    const float invt = s_invt;

    const int lane = tid & 31;
    const int wave = tid >> 5;    // 0..7
    const int half = lane >> 4;   // 0 or 1 (K sub-pair)
    const int r    = lane & 15;   // row within 16-row tile

    // Per-lane B fragments: broadcast target over N -> value depends only on K.
    v2f tb[16];
#pragma unroll
    for (int c = 0; c < 16; ++c) {
        tb[c].x = tgt[c * 4 + half * 2 + 0];
        tb[c].y = tgt[c * 4 + half * 2 + 1];
    }

    const int blockStart = blockIdx.x * ROWS_PER_BLOCK;

    for (int it = 0; it < ROWS_PER_BLOCK / 128; ++it) {   // 8 waves * 16 rows
        const int rowBase = blockStart + it * 128 + wave * 16;
        const int row     = rowBase + r;
        const bool valid  = row < V;
        const float* rp   = emb + (size_t)(valid ? row : 0) * 64;
        const v2f*   rp2  = (const v2f*)rp;

        // Prefetch next tile of this lane's row stream into L2.
        if (rowBase + 128 + r < V)
            __builtin_prefetch(emb + (size_t)(row + 128) * 64 + half * 2, 0, 1);

        v8f   acc   = {};
        float sumsq = 0.f;
#pragma unroll
        for (int c = 0; c < 16; ++c) {
            v2f a = rp2[c * 2 + half];                 // elems {c*4+2h, c*4+2h+1}
            sumsq = __builtin_fmaf(a.x, a.x, sumsq);
            sumsq = __builtin_fmaf(a.y, a.y, sumsq);
            acc = __builtin_amdgcn_wmma_f32_16x16x4_f32(
                false, a, false, tb[c], (short)0, acc, false, false);
        }

        // Row r's sum-of-squares lives in lanes r and r+16: combine.
        sumsq += __shfl_xor(sumsq, 16, 32);
        const float rs  = rsqrtf(sumsq) * invt;          // scale for row (lane&15)
        const float rs8 = __shfl(rs, lane + 8, 32);      // lanes 16..23 -> row 8+m

        // D layout: lanes <16 hold rows 0..7 in acc[0..7]; lanes >=16 rows 8..15.
        const int  m      = lane & 7;
        const bool writer = (lane < 8) || (lane >= 16 && lane < 24);
        if (writer) {
            const int rlocal = ((lane >= 16) ? 8 : 0) + m;
            const int grow   = rowBase + rlocal;
            const float scl  = (lane >= 16) ? rs8 : rs;
            const float val  = (grow < V) ? acc[m] * scl : -__builtin_inff();
            sims[it * 128 + wave * 16 + rlocal] = val;
        }
    }
    __syncthreads();

    // 11 rounds of block-wide argmax over the LDS sims tile.
    for (int k = 0; k < TOPK; ++k) {
        float bv = -__builtin_inff();
        int   bi = 0x7fffffff;
        for (int i = tid; i < ROWS_PER_BLOCK; i += 256) {
            float v = sims[i];
            if (better(v, i, bv, bi)) { bv = v; bi = i; }
        }
        redV[tid] = bv; redI[tid] = bi;
        __syncthreads();
        for (int s = 128; s > 0; s >>= 1) {
            if (tid < s) {
                if (better(redV[tid + s], redI[tid + s], redV[tid], redI[tid])) {
                    redV[tid] = redV[tid + s]; redI[tid] = redI[tid + s];
                }
            }
            __syncthreads();
        }
        if (tid == 0) {
            wsV[blockIdx.x * TOPK + k] = redV[0];
            wsI[blockIdx.x * TOPK + k] = blockStart + redI[0];
            sims[redI[0]] = -__builtin_inff();
        }
        __syncthreads();
    }
}

// ---------------------------------------------------------------------------
// Kernel B: merge per-block candidates, drop rank-0 (self), emit 10+10.
// ---------------------------------------------------------------------------
__global__ __launch_bounds__(256) void final_topk_kernel(
    const float* __restrict__ wsV, const int* __restrict__ wsI,
    int nCand, float* __restrict__ out)
{
    __shared__ float cv[MAXC];
    __shared__ int   ci[MAXC];
    __shared__ float redV[256];
    __shared__ int   redP[256];

    const int tid = threadIdx.x;
    for (int i = tid; i < nCand; i += 256) { cv[i] = wsV[i]; ci[i] = wsI[i]; }
    __syncthreads();

    for (int k = 0; k < TOPK; ++k) {
        float bv = -__builtin_inff();
        int   bp = 0;
        int   bg = 0x7fffffff;
        for (int i = tid; i < nCand; i += 256) {
            float v = cv[i];
            if (better(v, ci[i], bv, bg)) { bv = v; bp = i; bg = ci[i]; }
        }
        redV[tid] = bv; redP[tid] = bp;
        __syncthreads();
        for (int s = 128; s > 0; s >>= 1) {
            if (tid < s) {
                int   op = redP[tid + s];
                float ov = redV[tid + s];
                if (better(ov, ci[op], redV[tid], ci[redP[tid]])) {
                    redV[tid] = ov; redP[tid] = op;
                }
            }
            __syncthreads();
        }
        if (tid == 0) {
            const int pos = redP[0];
            if (k >= 1) {                       // rank 0 is the self-match
                out[k - 1]        = redV[0];
                out[10 + (k - 1)] = (float)ci[pos];
            }
            cv[pos] = -__builtin_inff();
        }
        __syncthreads();
    }
}

extern "C" void kernel_launch(void* const* d_in, const int* in_sizes, int n_in,
                              void* d_out, int out_size, void* d_ws, size_t ws_size,
                              hipStream_t stream) {
    (void)n_in; (void)out_size; (void)ws_size;
    const float* emb  = (const float*)d_in[0];
    const int*   uidp = (const int*)d_in[1];
    const int V = in_sizes[0] / 64;

    int nBlocks = (V + ROWS_PER_BLOCK - 1) / ROWS_PER_BLOCK;
    int nCand   = nBlocks * TOPK;
    if (nCand > MAXC) nCand = MAXC;   // safety clamp (never hit for V=1e6)

    float* wsV = (float*)d_ws;
    int*   wsI = (int*)(wsV + nBlocks * TOPK);

    sim_topk_block_kernel<<<nBlocks, 256, 0, stream>>>(emb, uidp, V, wsV, wsI);
    final_topk_kernel<<<1, 256, 0, stream>>>(wsV, wsI, nCand, (float*)d_out);
}